// RecurrentLIFNode_56831007261289
// MI455X (gfx1250) — compile-verified
//
#include <hip/hip_runtime.h>

typedef __attribute__((ext_vector_type(2))) float v2f;
typedef __attribute__((ext_vector_type(8))) float v8f;

// Problem constants (match reference: T=128, B=256, H=512, L=4, TAU=2, V_TH=1)
#define TT 128
#define BB 256
#define HH 512
#define LL 4

// One LIF layer step: y = inp @ W^T + b ; v = v_prev + (y - v_prev)/2 ;
// spike = (v >= 1) ; v -= spike. One wave per 16x16 output tile, fp32 WMMA.
__global__ __launch_bounds__(256) void lif_layer_step(
    const float* __restrict__ inp,    // [B,H] layer input (x_t or previous layer's v)
    const float* __restrict__ W,      // [H,H] row-major; y = inp @ W^T
    const float* __restrict__ bias,   // [H]
    float* __restrict__ vstate,       // [B,H] membrane state: read v_prev, write v
    float* __restrict__ spike_out)    // [B,H] spike output (last layer) or nullptr
{
    const int lane  = threadIdx.x & 31;
    const int wave  = threadIdx.x >> 5;
    const int gwave = blockIdx.x * 8 + wave;        // 0..511
    const int tileM = gwave & 15;                   // 16 tiles over B=256
    const int tileN = gwave >> 4;                   // 32 tiles over H=512
    const int half  = lane >> 4;                    // lane group (0: lanes 0-15, 1: 16-31)
    const int nl    = lane & 15;

    const int m_base = tileM * 16;
    const int n_base = tileN * 16;

    // ---- Hoisted epilogue operands: fully hidden under the GEMM loop. ----
    // C/D layout: VGPR r, lane -> element (m = r + 8*half, n = nl).
    const float bval = bias[n_base + nl];
    float* __restrict__ vrow = vstate + (size_t)(m_base + 8 * half) * HH + (n_base + nl);
    float vprev[8];
#pragma unroll
    for (int r = 0; r < 8; ++r)
        vprev[r] = vrow[r * HH];   // 8 loads in one clause, base + imm offsets

    // A (16x4 f32 per WMMA): lane supplies inp[m_base+nl][k0 + 2*half + j], j=0,1
    // B (4x16 f32): B[k][n] = W[n][k]  -> lane supplies W[n_base+nl][k0 + 2*half + j]
    const float* __restrict__ Aptr = inp + (size_t)(m_base + nl) * HH + 2 * half;
    const float* __restrict__ Bptr = W   + (size_t)(n_base + nl) * HH + 2 * half;

    v8f c = {};
#pragma unroll 16
    for (int k0 = 0; k0 < HH; k0 += 4) {
        v2f a = *(const v2f*)(Aptr + k0);
        v2f b = *(const v2f*)(Bptr + k0);
        // 8 args: (neg_a, A, neg_b, B, c_mod, C, reuse_a, reuse_b)
        c = __builtin_amdgcn_wmma_f32_16x16x4_f32(
                false, a, false, b, (short)0, c, false, false);
    }

    // ---- Fused LIF epilogue (no loads left here). ----
    float v[8], sp[8];
#pragma unroll
    for (int r = 0; r < 8; ++r) {
        const float y = c[r] + bval;
        float vv = vprev[r] + (y - vprev[r]) * 0.5f;     // neuronal charge, tau=2
        sp[r] = (vv >= 1.0f) ? 1.0f : 0.0f;              // heaviside fire
        v[r]  = vv - sp[r];                              // soft reset (V_TH = 1)
    }
#pragma unroll
    for (int r = 0; r < 8; ++r)
        vrow[r * HH] = v[r];
    if (spike_out != nullptr) {                           // single uniform branch
        float* __restrict__ srow =
            spike_out + (size_t)(m_base + 8 * half) * HH + (n_base + nl);
#pragma unroll
        for (int r = 0; r < 8; ++r)
            srow[r * HH] = sp[r];
    }
}

extern "C" void kernel_launch(void* const* d_in, const int* in_sizes, int n_in,
                              void* d_out, int out_size, void* d_ws, size_t ws_size,
                              hipStream_t stream) {
    const float* x       = (const float*)d_in[0];   // [T,B,H]
    const float* states0 = (const float*)d_in[1];   // [L,B,H]
    const float* Ws      = (const float*)d_in[2];   // [L,H,H]
    const float* bs      = (const float*)d_in[3];   // [L,H]

    float* spikes = (float*)d_out;                         // [T,B,H]
    float* vstate = spikes + (size_t)TT * BB * HH;         // [L,B,H] live state == final states

    // Initialize live membrane state from the input initial states.
    hipMemcpyAsync(vstate, states0, sizeof(float) * (size_t)LL * BB * HH,
                   hipMemcpyDeviceToDevice, stream);

    dim3 grid(64), block(256);   // 512 waves == 512 output tiles of 16x16
    for (int t = 0; t < TT; ++t) {
        for (int l = 0; l < LL; ++l) {
            const float* inp = (l == 0) ? (x + (size_t)t * BB * HH)
                                        : (vstate + (size_t)(l - 1) * BB * HH);
            lif_layer_step<<<grid, block, 0, stream>>>(
                inp,
                Ws + (size_t)l * HH * HH,
                bs + (size_t)l * HH,
                vstate + (size_t)l * BB * HH,
                (l == LL - 1) ? (spikes + (size_t)t * BB * HH) : nullptr);
        }
    }
}